// TabularAugmentedModel_38774964748597
// MI455X (gfx1250) — compile-verified
//
#include <hip/hip_runtime.h>
#include <math.h>

// ---------------------------------------------------------------------------
// Problem constants (match reference)
// ---------------------------------------------------------------------------
#define S_NVALS 1024
#define A_NVALS 32
#define R_NVALS 2
#define N_TRANS 262144
#define SA_ROWS (S_NVALS * A_NVALS)      // 32768 rows of length 1024
#define GATHER_BLOCKS (N_TRANS / 256)    // 1024 partial sums

typedef __attribute__((ext_vector_type(2))) float v2f;
typedef __attribute__((ext_vector_type(8))) float v8f;

// ---------------------------------------------------------------------------
// wave32 reductions
// ---------------------------------------------------------------------------
__device__ __forceinline__ float wave_max(float x) {
#pragma unroll
    for (int off = 16; off > 0; off >>= 1)
        x = fmaxf(x, __shfl_xor(x, off, 32));
    return x;
}

__device__ __forceinline__ float wave_sum(float x) {
#pragma unroll
    for (int off = 16; off > 0; off >>= 1)
        x += __shfl_xor(x, off, 32);
    return x;
}

// ---------------------------------------------------------------------------
// Kernel A: LSE over params_a_s rows (1024 x 32) and params_r_s rows (1024 x 2)
// Tiny tables (136 KB total) -> simplest correct form, one thread per state.
// ---------------------------------------------------------------------------
__global__ __launch_bounds__(256) void lse_small_kernel(
    const float* __restrict__ a_tab, const float* __restrict__ r_tab,
    float* __restrict__ lse_a, float* __restrict__ lse_r) {
    int s = blockIdx.x * blockDim.x + threadIdx.x;
    if (s >= S_NVALS) return;

    const float* pa = a_tab + s * A_NVALS;
    float m = -__builtin_inff();
#pragma unroll
    for (int i = 0; i < A_NVALS; ++i) m = fmaxf(m, pa[i]);
    float sum = 0.0f;
#pragma unroll
    for (int i = 0; i < A_NVALS; ++i) sum += __expf(pa[i] - m);
    lse_a[s] = m + __logf(sum);

    float r0 = r_tab[2 * s + 0];
    float r1 = r_tab[2 * s + 1];
    float mr = fmaxf(r0, r1);
    lse_r[s] = mr + __logf(__expf(r0 - mr) + __expf(r1 - mr));
}

// ---------------------------------------------------------------------------
// Kernel B (hot path, streams 128 MB once): LSE over each 1024-wide row of
// params_s_sa. One wave per row; row lives entirely in registers
// (8 x float4 per lane). Sum-of-exp is done on the XDL pipe with chained
// V_WMMA_F32_16X16X4_F32 against an all-ones B matrix: summation is
// order-agnostic, so each lane's 32 exp values are fed directly as the
// A-operand register pairs of 16 accumulating WMMAs (no data movement).
// The C accumulator then holds the full row sum replicated in every column;
// 8 register adds + one shfl_xor(16) finish the reduction exactly in f32.
// ---------------------------------------------------------------------------
__global__ __launch_bounds__(256) void lse_s_kernel(
    const float* __restrict__ tab, float* __restrict__ lse_s) {
    const int wave = threadIdx.x >> 5;
    const int lane = threadIdx.x & 31;
    const int row  = blockIdx.x * 8 + wave;   // grid sized exactly: 4096*8 = 32768

    const float4* p = reinterpret_cast<const float4*>(tab + (size_t)row * S_NVALS);

    // Coalesced: 8 x global_load_b128, 512B per instruction per wave.
    float4 v[8];
#pragma unroll
    for (int j = 0; j < 8; ++j) v[j] = p[lane + 32 * j];

    // Row max (register tree + 5 shuffle steps).
    float m = -__builtin_inff();
#pragma unroll
    for (int j = 0; j < 8; ++j)
        m = fmaxf(m, fmaxf(fmaxf(v[j].x, v[j].y), fmaxf(v[j].z, v[j].w)));
    m = wave_max(m);

    // exp(x - m) in place (v_exp_f32, co-executes with the WMMA chain below).
    float ev[32];
#pragma unroll
    for (int j = 0; j < 8; ++j) {
        ev[4 * j + 0] = __expf(v[j].x - m);
        ev[4 * j + 1] = __expf(v[j].y - m);
        ev[4 * j + 2] = __expf(v[j].z - m);
        ev[4 * j + 3] = __expf(v[j].w - m);
    }

    // Sum of 1024 exp values via 16 chained f32 WMMAs (exact f32 math).
    v2f bones;
    bones[0] = 1.0f;
    bones[1] = 1.0f;
    v8f c = {};
#pragma unroll
    for (int i = 0; i < 16; ++i) {
        v2f a;
        a[0] = ev[2 * i + 0];
        a[1] = ev[2 * i + 1];
        // (neg_a, A, neg_b, B, c_mod, C, reuse_a, reuse_b)
        c = __builtin_amdgcn_wmma_f32_16x16x4_f32(
            false, a, false, bones, (short)0, c, false, false);
    }
    // C[M][n] = rowsum_M summed over the 16 tiles (independent of n).
    float ssum = c[0] + c[1] + c[2] + c[3] + c[4] + c[5] + c[6] + c[7];
    ssum += __shfl_xor(ssum, 16, 32);   // rows 0-7 half + rows 8-15 half

    if (lane == 0) lse_s[row] = m + __logf(ssum);
}

// ---------------------------------------------------------------------------
// Kernel C: per-transition gather + per-block deterministic reduction.
// The 128 MB table is L2-resident (192 MB L2) after kernel B, so the random
// single-float gathers are cheap.
// ---------------------------------------------------------------------------
__global__ __launch_bounds__(256) void gather_kernel(
    const float* __restrict__ s_tab, const float* __restrict__ r_tab,
    const float* __restrict__ a_tab,
    const int* __restrict__ s_idx, const int* __restrict__ a_idx,
    const int* __restrict__ r_idx, const int* __restrict__ d_flag,
    const int* __restrict__ sn_idx,
    const float* __restrict__ lse_s, const float* __restrict__ lse_a,
    const float* __restrict__ lse_r, float* __restrict__ partials) {
    const int t = blockIdx.x * 256 + threadIdx.x;   // grid sized exactly

    const int s  = s_idx[t];
    const int a  = a_idx[t];
    const int r  = r_idx[t];
    const int d  = d_flag[t];
    const int sn = sn_idx[t];
    const int sa = s * A_NVALS + a;

    float lp = (a_tab[sa] - lse_a[s]) + (r_tab[s * R_NVALS + r] - lse_r[s]);
    float lpn = s_tab[(size_t)sa * S_NVALS + sn] - lse_s[sa];
    lp += d ? 0.0f : lpn;

    // Deterministic block reduction: wave shuffles -> LDS -> wave 0.
    float w = wave_sum(lp);
    __shared__ float smem[8];
    const int lane = threadIdx.x & 31;
    const int wv   = threadIdx.x >> 5;
    if (lane == 0) smem[wv] = w;
    __syncthreads();
    if (threadIdx.x < 32) {
        float x = (threadIdx.x < 8) ? smem[threadIdx.x] : 0.0f;
#pragma unroll
        for (int off = 4; off > 0; off >>= 1) x += __shfl_xor(x, off, 32);
        if (threadIdx.x == 0) partials[blockIdx.x] = x;
    }
}

// ---------------------------------------------------------------------------
// Kernel D: fixed-order final reduction of the 1024 block partials.
// ---------------------------------------------------------------------------
__global__ __launch_bounds__(256) void final_kernel(
    const float* __restrict__ partials, const int* __restrict__ bsz,
    float* __restrict__ out) {
    float x = 0.0f;
#pragma unroll
    for (int i = threadIdx.x; i < GATHER_BLOCKS; i += 256) x += partials[i];

    float w = wave_sum(x);
    __shared__ float smem[8];
    const int lane = threadIdx.x & 31;
    const int wv   = threadIdx.x >> 5;
    if (lane == 0) smem[wv] = w;
    __syncthreads();
    if (threadIdx.x == 0) {
        float total = 0.0f;
#pragma unroll
        for (int i = 0; i < 8; ++i) total += smem[i];
        out[0] = -total / (float)bsz[0];
    }
}

// ---------------------------------------------------------------------------
// Entry point
// ---------------------------------------------------------------------------
extern "C" void kernel_launch(void* const* d_in, const int* in_sizes, int n_in,
                              void* d_out, int out_size, void* d_ws, size_t ws_size,
                              hipStream_t stream) {
    (void)in_sizes; (void)n_in; (void)out_size; (void)ws_size;

    const float* params_s_sa = (const float*)d_in[0];  // (1024, 32, 1024)
    const float* params_r_s  = (const float*)d_in[1];  // (1024, 2)
    const float* params_a_s  = (const float*)d_in[2];  // (1024, 32)
    const int*   s_idx       = (const int*)d_in[3];
    const int*   a_idx       = (const int*)d_in[4];
    const int*   r_idx       = (const int*)d_in[5];
    const int*   d_flag      = (const int*)d_in[6];
    const int*   snext_idx   = (const int*)d_in[7];
    const int*   batch_size  = (const int*)d_in[8];

    float* out = (float*)d_out;

    // Workspace layout (floats): lse_s | lse_a | lse_r | block partials
    float* ws       = (float*)d_ws;
    float* lse_s    = ws;                       // 32768
    float* lse_a    = lse_s + SA_ROWS;          // 1024
    float* lse_r    = lse_a + S_NVALS;          // 1024
    float* partials = lse_r + S_NVALS;          // 1024

    lse_small_kernel<<<(S_NVALS + 255) / 256, 256, 0, stream>>>(
        params_a_s, params_r_s, lse_a, lse_r);

    lse_s_kernel<<<SA_ROWS / 8, 256, 0, stream>>>(params_s_sa, lse_s);

    gather_kernel<<<GATHER_BLOCKS, 256, 0, stream>>>(
        params_s_sa, params_r_s, params_a_s,
        s_idx, a_idx, r_idx, d_flag, snext_idx,
        lse_s, lse_a, lse_r, partials);

    final_kernel<<<1, 256, 0, stream>>>(partials, batch_size, out);
}